// Model_2628519985485
// MI455X (gfx1250) — compile-verified
//
#include <hip/hip_runtime.h>

// MI455X / gfx1250 implementation.
//
// Cost model (MI455X): HBM 23.3 TB/s, L2 192MB, wave32, WGP with 320KB LDS.
//  - W GEMM: 1.08 GFLOP over 16MB of A  -> f32 WMMA (16x16x4), ~us scale.
//  - t1 matvec: 16MB stream             -> bandwidth bound, ~us.
//  - scan: 8256 strictly sequential (2x 128x128 matvec + relu) steps
//    -> single-WGP LDS-resident kernel; issue-rate bound, dominant cost.

typedef __attribute__((ext_vector_type(2))) float v2f;
typedef __attribute__((ext_vector_type(8))) float v8f;

#define Mm 128     // m
#define Nn 32768   // N
#define Pp 128     // P
#define Kc 0.01f   // K

// --------------------------------------------------------------------------
// Kernel A: W[v,j] = dot(Fm_v, Fm_j) + bval^2, v in [0,129), j in [0,128)
// Fm_v = A[0,v] for v<128, Fm_128 = c.  One wave32 per 16x16 output tile,
// accumulating with v_wmma_f32_16x16x4_f32 over K=32768.
// A fragment (ISA 7.12.2, 32-bit A 16x4): lane L holds row M=L&15,
//   VGPR0/1 = K = 2*(L>=16) + {0,1}  -> one contiguous float2 per step.
// B fragment ("row striped across lanes within one VGPR"): lane L holds
//   col N=L&15, VGPR0/1 = B[K = (L>=16) + {0,2}][N] -> two stride-2 scalars.
// Since B[k][n] = X[rowN][k], both operands are loads from X rows.
// --------------------------------------------------------------------------
__global__ __launch_bounds__(32) void w_gemm_kernel(const float* __restrict__ A,
                                                    const float* __restrict__ c,
                                                    const float* __restrict__ b,
                                                    float* __restrict__ W) {
  const int lane  = threadIdx.x;   // wave32
  const int ntile = blockIdx.x;    // 0..7   (j tiles)
  const int mtile = blockIdx.y;    // 0..8   (v tiles; tile 8 holds only v=128)
  const int mr = lane & 15;
  const int hk = lane >> 4;        // K half: 0 or 1

  const int rowM = mtile * 16 + mr;
  const int rowN = ntile * 16 + mr;                    // always < 128
  const float* pA = (rowM < Mm) ? (A + (size_t)rowM * Nn) : c;  // v=128 -> c; v>128 dummy
  const float* pB = A + (size_t)rowN * Nn;

  v8f acc = {0.f, 0.f, 0.f, 0.f, 0.f, 0.f, 0.f, 0.f};
  #pragma unroll 8
  for (int k = 0; k < Nn; k += 4) {
    v2f a = *(const v2f*)(pA + k + 2 * hk);   // A[m][k+2hk], A[m][k+2hk+1]
    v2f bb;
    bb.x = pB[k + hk];                        // B[hk  ][n]
    bb.y = pB[k + 2 + hk];                    // B[hk+2][n]
    acc = __builtin_amdgcn_wmma_f32_16x16x4_f32(
        /*neg_a=*/false, a, /*neg_b=*/false, bb,
        /*c_mod=*/(short)0, acc, /*reuse_a=*/false, /*reuse_b=*/false);
  }

  // Fb's extra column (value bval in every row) contributes bval^2 to each W.
  const float bv  = b[1];          // b[0,1,0]
  const float bsq = bv * bv;

  // C/D layout: VGPR r: lanes 0-15 -> (M=r, N=lane), lanes 16-31 -> (M=8+r, N=lane-16)
  const int n  = lane & 15;
  const int mb = (lane >> 4) * 8;
  if (mtile < 8) {                       // uniform branch: all 8 rows valid
    #pragma unroll
    for (int r = 0; r < 8; ++r)
      W[(mtile * 16 + mb + r) * Pp + ntile * 16 + n] = acc[r] + bsq;
  } else if (mb == 0) {                  // tile 8: only v=128 (M=0, lanes 0-15)
    W[Mm * Pp + ntile * 16 + n] = acc[0] + bsq;
  }
}

// --------------------------------------------------------------------------
// Kernel B: t1[p] = K * dot(theta1[p], A[z])   (bandwidth-bound reduction)
// --------------------------------------------------------------------------
__global__ __launch_bounds__(256) void t1_kernel(const float* __restrict__ theta1,
                                                 const float* __restrict__ A,
                                                 const int* __restrict__ zp,
                                                 float* __restrict__ t1) {
  __shared__ float red[256];
  const int p   = blockIdx.x;
  const int tid = threadIdx.x;
  const float* az = A + (size_t)zp[0] * Nn;
  const float* th = theta1 + (size_t)p * Nn;
  float s = 0.f;
  for (int k = tid * 4; k < Nn; k += 256 * 4) {
    float4 a = *(const float4*)(az + k);
    float4 t = *(const float4*)(th + k);
    s += a.x * t.x + a.y * t.y + a.z * t.z + a.w * t.w;
  }
  red[tid] = s;
  __syncthreads();
  for (int off = 128; off > 0; off >>= 1) {
    if (tid < off) red[tid] += red[tid + off];
    __syncthreads();
  }
  if (tid == 0) t1[p] = Kc * red[0];
}

// --------------------------------------------------------------------------
// Kernel C: Bmat[v][p] = t1[p] + T3[v,p] + T3c[v,p]
// T3[v,p] = K * sum_{j!=v} relu(th3[p]*W[v,j])
//         = K * ( max(th3,0)*Spos[v] + max(-th3,0)*Sneg[v] )
// T3c[v,p] = K * relu(theta3c[p] * wcv[ulast[v]]),  wcv[j] = W[128,j] - bval^2
// --------------------------------------------------------------------------
__global__ __launch_bounds__(128) void bmat_kernel(const float* __restrict__ W,
                                                   const float* __restrict__ t1,
                                                   const float* __restrict__ theta3,
                                                   const float* __restrict__ theta3c,
                                                   const float* __restrict__ b,
                                                   float* __restrict__ Bm) {
  __shared__ float rp[128];
  __shared__ float rn[128];
  const int v   = blockIdx.x;    // 0..128
  const int tid = threadIdx.x;   // doubles as j (reduction) and p (output)

  const float w = W[v * Pp + tid];
  rp[tid] = (tid == v) ? 0.f : fmaxf(w, 0.f);
  rn[tid] = (tid == v) ? 0.f : fmaxf(-w, 0.f);
  __syncthreads();
  for (int off = 64; off > 0; off >>= 1) {
    if (tid < off) { rp[tid] += rp[tid + off]; rn[tid] += rn[tid + off]; }
    __syncthreads();
  }
  const float Spos = rp[0], Sneg = rn[0];

  const float th3 = theta3[tid];
  const float T3  = Kc * (fmaxf(th3, 0.f) * Spos + fmaxf(-th3, 0.f) * Sneg);

  const float bv = b[1];
  const int   ul = (v == Mm - 1) ? (Mm - 2) : (Mm - 1);
  const float wcv = W[Mm * Pp + ul] - bv * bv;
  const float T3c = Kc * fmaxf(theta3c[tid] * wcv, 0.f);

  Bm[v * Pp + tid] = t1[tid] + T3 + T3c;
}

// --------------------------------------------------------------------------
// Kernel D: the sequential scan (64 timesteps x 129 node steps), single
// workgroup of 256 threads (8 wave32 on one WGP), ~202KB LDS resident.
//
// theta2/theta2c are interleaved into one matrix sT so the inner loop is
// pure b128 traffic: row i, half h holds 32 float4 groups alternating
//   [t2(jj..jj+3)], [t2c(jj..jj+3)], ...
// and the combined vector sXY holds matching groups [x(jj..jj+3)],
// [y(jj..jj+3)], ... (y = mu[:,128]).  Per step each thread issues
// 64 ds_load_b128 + 128 FMAs (vs 128 b64 before).  Row stride 260 floats
// keeps every group 16B-aligned.
// --------------------------------------------------------------------------
#define TROW 260
__global__ __launch_bounds__(256) void scan_kernel(
    const float* __restrict__ theta2, const float* __restrict__ theta2c,
    const float* __restrict__ mu0, const float* __restrict__ Bm,
    const float* __restrict__ theta4a, const float* __restrict__ theta4b,
    const int* __restrict__ zp, float* __restrict__ out) {
  __shared__ __align__(16) float sT  [128 * TROW];   // 133120 B
  __shared__ __align__(16) float sMu [129 * 128];    //  66048 B, [v*128 + p]
  __shared__ __align__(16) float sXY [256];          // interleaved x/y
  __shared__ float sS  [128];
  __shared__ float sPart[256];

  const int tid = threadIdx.x;

  // Stage + interleave theta2/theta2c into LDS.
  for (int idx = tid; idx < 128 * 128; idx += 256) {
    const int i = idx >> 7, j = idx & 127;
    const int h = j >> 6, jj = j & 63;
    const int g = jj >> 2, r = jj & 3;
    const int base = i * TROW + h * 128 + g * 8 + r;
    sT[base]     = theta2 [idx];
    sT[base + 4] = theta2c[idx];
  }
  // mu transposed so a node column is contiguous: sMu[v*128+p] = mu0[p,v].
  for (int idx = tid; idx < 129 * 128; idx += 256) {
    const int v = idx >> 7, p = idx & 127;
    sMu[idx] = mu0[p * 129 + v];
  }
  __syncthreads();

  // xyIdx: where column p of x lives in sXY (y lives at +4).
  const int xp = tid & 127;
  const int xyIdx = (xp >> 6) * 128 + ((xp & 63) >> 2) * 8 + (xp & 3);

  if (tid < 128) {
    float s = 0.f;
    for (int v = 0; v < 128; ++v) s += sMu[v * 128 + tid];
    sS[tid] = s;
    sXY[xyIdx]     = s - sMu[tid];            // x for v=0
    sXY[xyIdx + 4] = sMu[128 * 128 + tid];    // y = mu[:,128]
  }

  const int i = tid >> 1, h = tid & 1;
  const float* tRow = &sT[i * TROW + h * 128];
  const float* xyH  = &sXY[h * 128];

  for (int t = 0; t < 64; ++t) {
    for (int v = 0; v < 129; ++v) {
      __syncthreads();  // sXY ready; previous update visible
      // Prefetch this step's Bmat row from L2; latency hidden by the matvec.
      const float bm = (tid < 128) ? Bm[v * 128 + tid] : 0.f;
      float acc = 0.f;
      #pragma unroll
      for (int q = 0; q < 32; ++q) {
        const float4 tq = *(const float4*)(tRow + 4 * q);
        const float4 xq = *(const float4*)(xyH  + 4 * q);
        acc += tq.x * xq.x + tq.y * xq.y + tq.z * xq.z + tq.w * xq.w;
      }
      sPart[tid] = acc;
      __syncthreads();
      if (tid < 128) {
        const float pre = bm + Kc * (sPart[2 * tid] + sPart[2 * tid + 1]);
        const float nw  = fmaxf(pre, 0.f);
        const float old = sMu[v * 128 + tid];
        sMu[v * 128 + tid] = nw;
        if (v < 128) sS[tid] += nw - old;
        else         sXY[xyIdx + 4] = nw;     // refresh y = mu[:,128]
        // Prepare x for the next node step (wraps to v=0 of next timestep).
        const int vn = (v == 128) ? 0 : (v + 1);
        const float mvn = sMu[vn * 128 + tid];
        sXY[xyIdx] = sS[tid] - ((vn < 128) ? mvn : 0.f);
      }
    }
  }
  __syncthreads();

  // feat = concat(theta4a @ S, theta4b @ mu[:, z])
  const int z = zp[0];
  if (tid < 128) {
    float s = 0.f;
    for (int j = 0; j < 128; ++j) s += theta4a[tid * 128 + j] * sS[j];
    out[tid] = s;
  } else {
    const int o = tid - 128;
    float s = 0.f;
    for (int j = 0; j < 128; ++j) s += theta4b[o * 128 + j] * sMu[z * 128 + j];
    out[128 + o] = s;
  }
}

// --------------------------------------------------------------------------
extern "C" void kernel_launch(void* const* d_in, const int* in_sizes, int n_in,
                              void* d_out, int out_size, void* d_ws, size_t ws_size,
                              hipStream_t stream) {
  (void)in_sizes; (void)n_in; (void)out_size; (void)ws_size;
  const float* A    = (const float*)d_in[0];   // (1,128,32768)
  const float* b    = (const float*)d_in[1];   // (1,2,1)
  const float* c    = (const float*)d_in[2];   // (1,32768)
  const float* mu0  = (const float*)d_in[3];   // (128,129)
  const float* th1  = (const float*)d_in[4];   // (128,32768)
  const float* th2  = (const float*)d_in[5];   // (128,128)
  const float* th2c = (const float*)d_in[6];   // (128,128)
  const float* th3  = (const float*)d_in[7];   // (128,1)
  const float* th3c = (const float*)d_in[8];   // (128,1)
  const float* th4a = (const float*)d_in[9];   // (128,128)
  const float* th4b = (const float*)d_in[10];  // (128,128)
  const int*   z    = (const int*)d_in[11];    // scalar

  float* W  = (float*)d_ws;        // 129*128
  float* t1 = W + 129 * 128;       // 128
  float* Bm = t1 + 128;            // 129*128   (total ws: ~133KB)
  float* out = (float*)d_out;      // 256 f32

  w_gemm_kernel<<<dim3(8, 9), 32, 0, stream>>>(A, c, b, W);
  t1_kernel<<<128, 256, 0, stream>>>(th1, A, z, t1);
  bmat_kernel<<<129, 128, 0, stream>>>(W, t1, th3, th3c, b, Bm);
  scan_kernel<<<1, 256, 0, stream>>>(th2, th2c, mu0, Bm, th4a, th4b, z, out);
}